// ClassCapsules_55362128445785
// MI455X (gfx1250) — compile-verified
//
#include <hip/hip_runtime.h>
#include <math.h>

typedef __attribute__((ext_vector_type(2))) float v2f;
typedef __attribute__((ext_vector_type(8))) float v8f;

#define B_TOT   256
#define N_IN    1152
#define D_IN    8
#define N_CLS   10
#define D_OUT   16
#define NCOL    160          // N_CLS * D_OUT flattened column dim
#define BCHUNK  64           // batch chunk resident in ws/L2 (47 MB)
#define BT      (BCHUNK/16)  // 16-row batch tiles per chunk

// -------------------------------------------------------------------------
// K1: u_hat[bb,i,c,o] = sum_d W[i,c,o,d] * x[b0+bb,i,d]
// One wave per (i, batch-tile). f32 WMMA 16x16x4, K=8 via two WMMAs.
// A-frag (16x4 f32): lane L: M=L&15, half=L>>4; v0=K(2*half), v1=K(2*half+1)
// B-frag (4x16 f32): lane L: N=L&15, half=L>>4; same K striping
// D (16x16 f32, 8 VGPRs): reg r, lane L -> M = r + 8*(L>>4), N = L&15
// -------------------------------------------------------------------------
__global__ __launch_bounds__(256) void caps_uhat_wmma(
    const float* __restrict__ x, const float* __restrict__ W,
    float* __restrict__ uh, int b0)
{
    const int lane  = threadIdx.x & 31;
    const int wave  = threadIdx.x >> 5;
    const int i     = blockIdx.x * 8 + wave;   // 0..1151
    const int btile = blockIdx.y;              // 0..BT-1
    const int nl    = lane & 15;               // A: row M / B: col N
    const int kh    = lane >> 4;               // K half select

    // A fragments from x[bg, i, 0..7] (reused across all 10 column tiles)
    const int bg = b0 + btile * 16 + nl;
    const float* xr = x + ((size_t)bg * N_IN + i) * D_IN;
    v2f a0, a1;
    a0.x = xr[2 * kh + 0]; a0.y = xr[2 * kh + 1];
    a1.x = xr[4 + 2 * kh + 0]; a1.y = xr[4 + 2 * kh + 1];

    const int bb_base = btile * 16;
    for (int ct = 0; ct < N_CLS; ++ct) {
        // B fragment: B[k][n] = W[i, ct, n, k]
        const float* Wr = W + (((size_t)i * N_CLS + ct) * D_OUT + nl) * D_IN;
        v2f bf0, bf1;
        bf0.x = Wr[2 * kh + 0]; bf0.y = Wr[2 * kh + 1];
        bf1.x = Wr[4 + 2 * kh + 0]; bf1.y = Wr[4 + 2 * kh + 1];

        v8f acc = {};
        acc = __builtin_amdgcn_wmma_f32_16x16x4_f32(false, a0, false, bf0,
                                                    (short)0, acc, false, false);
        acc = __builtin_amdgcn_wmma_f32_16x16x4_f32(false, a1, false, bf1,
                                                    (short)0, acc, false, false);
        // scatter D tile: rows = batch-in-chunk, cols = (ct*16 + n)
        #pragma unroll
        for (int r = 0; r < 8; ++r) {
            const int bb = bb_base + r + 8 * kh;
            uh[((size_t)bb * N_IN + i) * NCOL + ct * 16 + nl] = acc[r];
        }
    }
}

// -------------------------------------------------------------------------
// K2: fused dynamic routing, one workgroup per batch element of the chunk.
// b and c live in LDS across all 3 iterations; u_hat read from (L2-hot) ws.
// -------------------------------------------------------------------------
__global__ __launch_bounds__(320) void caps_route(
    const float* __restrict__ uh, float* __restrict__ out, int b0)
{
    __shared__ float bvec[N_IN][N_CLS];   // routing logits  (~45 KB)
    __shared__ float cmat[N_IN][N_CLS];   // softmax coeffs  (~45 KB)
    __shared__ float sbuf[NCOL];
    __shared__ float vbuf[NCOL];

    const int t  = threadIdx.x;           // 0..319
    const int bb = blockIdx.x;            // batch element within chunk
    const float* uhb = uh + (size_t)bb * N_IN * NCOL;

    // b = 0
    for (int idx = t; idx < N_IN * N_CLS; idx += 320)
        (&bvec[0][0])[idx] = 0.0f;
    __syncthreads();

    for (int iter = 0; iter < 3; ++iter) {
        // --- A: c = softmax(b) over classes (per input capsule) ---
        for (int i = t; i < N_IN; i += 320) {
            float mx = bvec[i][0];
            #pragma unroll
            for (int c = 1; c < N_CLS; ++c) mx = fmaxf(mx, bvec[i][c]);
            float e[N_CLS], sum = 0.0f;
            #pragma unroll
            for (int c = 0; c < N_CLS; ++c) {
                e[c] = __expf(bvec[i][c] - mx);
                sum += e[c];
            }
            const float inv = 1.0f / sum;
            #pragma unroll
            for (int c = 0; c < N_CLS; ++c) cmat[i][c] = e[c] * inv;
        }
        __syncthreads();

        // --- B: s[c,o] = sum_i c[i,c] * u_hat[i,c,o]  (two i-partials) ---
        float acc = 0.0f;
        const int co   = t % NCOL;        // 0..159
        const int part = t / NCOL;        // 0 or 1 (t<320)
        if (t < 2 * NCOL) {
            const int c  = co >> 4;
            const int i0 = part * (N_IN / 2);
            const int i1 = i0 + (N_IN / 2);
            #pragma unroll 4
            for (int i = i0; i < i1; ++i)
                acc += cmat[i][c] * uhb[(size_t)i * NCOL + co];
        }
        if (t < NCOL) sbuf[t] = acc;
        __syncthreads();
        if (t >= NCOL && t < 2 * NCOL) sbuf[co] += acc;  // unique co per thread
        __syncthreads();

        // --- C: v = squash(s) ---
        if (t < N_CLS) {
            float sq = 0.0f;
            #pragma unroll
            for (int o = 0; o < D_OUT; ++o) {
                const float sv = sbuf[t * 16 + o];
                sq += sv * sv;
            }
            const float scale = sq / ((1.0f + sq) * (sqrtf(sq) + 1e-8f));
            #pragma unroll
            for (int o = 0; o < D_OUT; ++o)
                vbuf[t * 16 + o] = scale * sbuf[t * 16 + o];
        }
        __syncthreads();

        // --- D: b += <u_hat, v>  (skipped on final iteration) ---
        if (iter < 2) {
            for (int i = t; i < N_IN; i += 320) {
                const float* up = uhb + (size_t)i * NCOL;
                #pragma unroll
                for (int c = 0; c < N_CLS; ++c) {
                    float agr = 0.0f;
                    #pragma unroll
                    for (int o = 0; o < D_OUT; ++o)
                        agr += up[c * 16 + o] * vbuf[c * 16 + o];
                    bvec[i][c] += agr;
                }
            }
            __syncthreads();
        }
    }

    if (t < NCOL) out[(size_t)(b0 + bb) * NCOL + t] = vbuf[t];
}

// -------------------------------------------------------------------------
extern "C" void kernel_launch(void* const* d_in, const int* in_sizes, int n_in,
                              void* d_out, int out_size, void* d_ws, size_t ws_size,
                              hipStream_t stream) {
    (void)in_sizes; (void)n_in; (void)out_size; (void)ws_size;
    const float* x = (const float*)d_in[0];   // [256,1152,8]
    const float* W = (const float*)d_in[1];   // [1,1152,10,16,8]
    float* out = (float*)d_out;               // [256,10,16]
    float* uh  = (float*)d_ws;                // chunk u_hat: [64,1152,160] = 47 MB

    for (int chunk = 0; chunk < B_TOT / BCHUNK; ++chunk) {
        const int b0 = chunk * BCHUNK;
        caps_uhat_wmma<<<dim3(N_IN / 8, BT), 256, 0, stream>>>(x, W, uh, b0);
        caps_route<<<BCHUNK, 320, 0, stream>>>(uh, out, b0);
    }
}